// Quantumnet_9165460210267
// MI455X (gfx1250) — compile-verified
//
#include <hip/hip_runtime.h>
#include <hip/hip_bf16.h>

typedef __attribute__((ext_vector_type(2))) float v2f;
typedef __attribute__((ext_vector_type(8))) float v8f;

#define INV_SQRT2_F 0.70710678118654752440f
#define QPI_QUART   0.78539816339744830962f  // pi/4

// ---------------- 4-qubit gates (fully unrolled, register-resident) ---------
// state index: i = b_q0*8 + b_q1*4 + b_q2*2 + b_q3  (qubit w -> bit (8>>w))

template<int W>
__device__ __forceinline__ void apply_h(float* st) {
  constexpr int b = 8 >> W;
#pragma unroll
  for (int i = 0; i < 16; ++i) {
    if (!(i & b)) {
      float s0 = st[i], s1 = st[i + b];
      st[i]     = (s0 + s1) * INV_SQRT2_F;
      st[i + b] = (s0 - s1) * INV_SQRT2_F;
    }
  }
}

template<int W>
__device__ __forceinline__ void apply_ry(float* st, float c, float s) {
  constexpr int b = 8 >> W;
#pragma unroll
  for (int i = 0; i < 16; ++i) {
    if (!(i & b)) {
      float s0 = st[i], s1 = st[i + b];
      st[i]     = c * s0 - s * s1;
      st[i + b] = s * s0 + c * s1;
    }
  }
}

template<int C, int T>
__device__ __forceinline__ void apply_cnot(float* st) {
  constexpr int bc = 8 >> C, bt = 8 >> T;
#pragma unroll
  for (int i = 0; i < 16; ++i) {
    if ((i & bc) && !(i & bt)) {
      float t = st[i]; st[i] = st[i + bt]; st[i + bt] = t;
    }
  }
}

// tanh via hardware exp; exact limits at +-inf
__device__ __forceinline__ float fast_tanh(float x) {
  float e = __expf(2.0f * x);
  return 1.0f - __fdividef(2.0f, e + 1.0f);
}

// ---------------- fused kernel ----------------------------------------------
// block = 256 threads (8 waves). wave w owns rows [blk*256 + w*32, +32).
// K=512 in 32 chunks of 16 cols; async HBM->LDS double buffer; WMMA f32 16x16x4.

#define WSTRIDE 528   // pre_W LDS row stride (floats): bank skew
#define ASTRIDE 20    // A-tile LDS row stride (floats): mult of 4, bank-skewed
#define CHUNK   16
#define NCHUNK  32
#define BUFSZ   (32 * ASTRIDE)          // floats per buffer (640)
#define BUFBYTES (BUFSZ * 4)            // 2560

__global__ __launch_bounds__(256) void qnet_fused_kernel(
    const float* __restrict__ X,      // [B,512]
    const float* __restrict__ preW,   // [4,512]
    const float* __restrict__ preB,   // [4]
    const float* __restrict__ qp,     // [60] = [15,4]
    const float* __restrict__ postW,  // [2,4]
    const float* __restrict__ postB,  // [2]
    float* __restrict__ out,          // [B,2]
    int Bn)
{
  __shared__ float sW[4 * WSTRIDE];       // 8448 B
  __shared__ float sA[8][2][BUFSZ];       // 40960 B

  const int tid   = threadIdx.x;
  const int wv    = tid >> 5;
  const int lane  = tid & 31;
  const int n     = lane & 15;      // N (C col) / M (A-frag row)
  const int half  = lane >> 4;      // K sub-pair selector within frag
  const int rowWave = blockIdx.x * 256 + wv * 32;

  // ---- async copy setup: lane -> (row lane>>2 (+8 per instr), 4 cols) -----
  // per instruction: 8 rows x 4 lanes x 16B (64B/row); 4 instrs per 2KB chunk
  const uint32_t ldsLane =
      (uint32_t)(uintptr_t)(&sA[wv][0][0]) +
      (uint32_t)(((lane >> 2) * ASTRIDE + (lane & 3) * 4) * 4);
  const int32_t voffLane =
      (int32_t)(((rowWave + (lane >> 2)) * 512 + (lane & 3) * 4) * 4);
  const uint64_t sbase = (uint64_t)(uintptr_t)X;

#define ISSUE_CHUNK(cc)                                                     \
  {                                                                         \
    uint32_t _d = ldsLane + (uint32_t)(((cc) & 1) * BUFBYTES);              \
    int32_t  _v = voffLane + (cc) * (CHUNK * 4);                            \
    _Pragma("unroll")                                                       \
    for (int _i = 0; _i < 4; ++_i) {                                        \
      asm volatile("global_load_async_to_lds_b128 %0, %1, %2"               \
                   :: "v"(_d + (uint32_t)(_i * 8 * ASTRIDE * 4)),           \
                      "v"(_v + _i * 8 * 512 * 4),                           \
                      "s"(sbase)                                            \
                   : "memory");                                             \
    }                                                                       \
  }

  // kick off chunk 0 before touching sW
  ISSUE_CHUNK(0);

  // ---- cooperative load of pre_W into padded LDS --------------------------
  for (int idx = tid; idx < 2048; idx += 256) {
    int r = idx >> 9, c = idx & 511;
    sW[r * WSTRIDE + c] = preW[idx];
  }
  __syncthreads();   // sW visible to all waves

  v8f acc0 = {0.f,0.f,0.f,0.f,0.f,0.f,0.f,0.f};
  v8f acc1 = {0.f,0.f,0.f,0.f,0.f,0.f,0.f,0.f};

  for (int c = 0; c < NCHUNK; ++c) {
    if (c + 1 < NCHUNK) {
      ISSUE_CHUNK(c + 1);
      // chunk c's 4 async ops retire in order -> wait cnt<=4
      asm volatile("s_wait_asynccnt 0x4" ::: "memory");
    } else {
      asm volatile("s_wait_asynccnt 0x0" ::: "memory");
    }
    const float* buf = &sA[wv][c & 1][0];
#pragma unroll
    for (int kk = 0; kk < 4; ++kk) {
      const int ko = kk * 4 + half * 2;
      // B frag 4x16: lanes n>=4 replicate col n&3 (C cols 4..15 unused)
      v2f bf = *(const v2f*)(sW + (n & 3) * WSTRIDE + c * CHUNK + ko);
      v2f a0 = *(const v2f*)(buf + n * ASTRIDE + ko);
      v2f a1 = *(const v2f*)(buf + (16 + n) * ASTRIDE + ko);
      acc0 = __builtin_amdgcn_wmma_f32_16x16x4_f32(
                 false, a0, false, bf, (short)0, acc0, false, false);
      acc1 = __builtin_amdgcn_wmma_f32_16x16x4_f32(
                 false, a1, false, bf, (short)0, acc1, false, false);
    }
  }

  // ---- transpose C through LDS: lane <- its own sample's 4 pre-activations
  // C layout: acc elem i of lane (half,n) = C[row half*8+i][col n]
  float* cbuf = &sA[wv][0][0];
  if (n < 4) {
#pragma unroll
    for (int i = 0; i < 8; ++i) {
      cbuf[(half * 8 + i) * ASTRIDE + n]      = acc0[i];
      cbuf[(16 + half * 8 + i) * ASTRIDE + n] = acc1[i];
    }
  }
  __syncthreads();

  float4 pre = *(const float4*)(cbuf + lane * ASTRIDE);  // 80B stride: 16B ok

  // ---- tanh -> half angles for the input RY layer -------------------------
  float h0 = fast_tanh(pre.x + preB[0]) * QPI_QUART;  // theta/2 = tanh*pi/4
  float h1 = fast_tanh(pre.y + preB[1]) * QPI_QUART;
  float h2 = fast_tanh(pre.z + preB[2]) * QPI_QUART;
  float h3 = fast_tanh(pre.w + preB[3]) * QPI_QUART;
  float ci0 = __cosf(h0), si0 = __sinf(h0);
  float ci1 = __cosf(h1), si1 = __sinf(h1);
  float ci2 = __cosf(h2), si2 = __sinf(h2);
  float ci3 = __cosf(h3), si3 = __sinf(h3);

  // shared layer weights (rows 1..6 of [15,4]); uniform across lanes
  float cw[6][4], sw[6][4];
#pragma unroll
  for (int k = 0; k < 6; ++k) {
#pragma unroll
    for (int w = 0; w < 4; ++w) {
      float h = qp[(k + 1) * 4 + w] * 0.5f;
      cw[k][w] = __cosf(h);
      sw[k][w] = __sinf(h);
    }
  }

  // ---- 16-amplitude circuit sim in registers ------------------------------
  float st[16];
#pragma unroll
  for (int i = 0; i < 16; ++i) st[i] = 0.f;
  st[0] = 1.f;

  apply_h<0>(st); apply_h<1>(st); apply_h<2>(st); apply_h<3>(st);
  apply_ry<0>(st, ci0, si0);
  apply_ry<1>(st, ci1, si1);
  apply_ry<2>(st, ci2, si2);
  apply_ry<3>(st, ci3, si3);

#pragma unroll
  for (int k = 0; k < 6; ++k) {
    apply_cnot<0,1>(st);
    apply_cnot<2,3>(st);
    apply_cnot<1,2>(st);
    apply_ry<0>(st, cw[k][0], sw[k][0]);
    apply_ry<1>(st, cw[k][1], sw[k][1]);
    apply_ry<2>(st, cw[k][2], sw[k][2]);
    apply_ry<3>(st, cw[k][3], sw[k][3]);
  }

  // ---- PauliZ expectations ------------------------------------------------
  float z0 = 0.f, z1 = 0.f, z2 = 0.f, z3 = 0.f;
#pragma unroll
  for (int i = 0; i < 16; ++i) {
    float p = st[i] * st[i];
    z0 += (i & 8) ? -p : p;
    z1 += (i & 4) ? -p : p;
    z2 += (i & 2) ? -p : p;
    z3 += (i & 1) ? -p : p;
  }

  // ---- post linear + coalesced b64 store ----------------------------------
  float o0 = z0*postW[0] + z1*postW[1] + z2*postW[2] + z3*postW[3] + postB[0];
  float o1 = z0*postW[4] + z1*postW[5] + z2*postW[6] + z3*postW[7] + postB[1];
  float2 ov; ov.x = o0; ov.y = o1;
  ((float2*)out)[rowWave + lane] = ov;
}

extern "C" void kernel_launch(void* const* d_in, const int* in_sizes, int n_in,
                              void* d_out, int out_size, void* d_ws, size_t ws_size,
                              hipStream_t stream) {
  const float* X     = (const float*)d_in[0];
  const float* preW  = (const float*)d_in[1];
  const float* preB  = (const float*)d_in[2];
  const float* qp    = (const float*)d_in[3];
  const float* postW = (const float*)d_in[4];
  const float* postB = (const float*)d_in[5];
  float* out = (float*)d_out;

  const int Bn = in_sizes[0] / 512;     // 131072
  const int grid = Bn / 256;            // 512 blocks of 256 threads
  qnet_fused_kernel<<<grid, 256, 0, stream>>>(X, preW, preB, qp, postW, postB,
                                              out, Bn);
}